// GNN_11751030522159
// MI455X (gfx1250) — compile-verified
//
#include <hip/hip_runtime.h>

// ---------------------------------------------------------------------------
// GCN (2x GCNConv + global_add_pool + linear) for MI455X (gfx1250, wave32).
//
// Roofline: edge phase dominates (~140B/edge x 3.2M x 2 convs ~ 0.9GB ->
// ~40us @ 23.3TB/s). h (6.4MB) fits in the 192MB L2, so random gathers and
// the fp32 atomic scatters resolve at L2. norm[] is computed once and reused
// by both convs. Dense 16-wide feature transforms use V_WMMA_F32_16X16X4_F32
// (exact f32 math; one 16-node x 16-feature tile per wave32) with branchless
// fragment loads (weights pre-transposed so B pairs load as b64) and a
// wave-uniform full-tile store fast path (no per-lane exec juggling).
// ---------------------------------------------------------------------------

#define NF 16  // feature width of both conv layers

typedef __attribute__((ext_vector_type(2)))  float v2f;
typedef __attribute__((ext_vector_type(8)))  float v8f;

__device__ __forceinline__ void atomAddF(float* p, float v) {
#if defined(__AMDGCN__)
  unsafeAtomicAdd(p, v);   // -> global_atomic_add_f32 (HW fp atomic, L2)
#else
  atomicAdd(p, v);
#endif
}

// ---- degree / norm precompute ---------------------------------------------

__global__ void init_deg_k(float* __restrict__ deg, int n) {
  int i = blockIdx.x * blockDim.x + threadIdx.x;
  if (i < n) deg[i] = 1.0f;  // self-loop weight
}

__global__ void deg_scatter_k(const int* __restrict__ col,
                              const float* __restrict__ ew,
                              float* __restrict__ deg, int n_edges) {
  int e = blockIdx.x * blockDim.x + threadIdx.x;
  if (e < n_edges) atomAddF(&deg[col[e]], ew[e]);
}

__global__ void dinv_k(float* __restrict__ deg, int n) {
  int i = blockIdx.x * blockDim.x + threadIdx.x;
  if (i < n) {
    float d = deg[i];
    deg[i] = (d > 0.0f) ? rsqrtf(d) : 0.0f;  // deg buffer becomes dinv
  }
}

__global__ void norm_k(const int* __restrict__ row, const int* __restrict__ col,
                       const float* __restrict__ ew, const float* __restrict__ dinv,
                       float* __restrict__ norm, int n_edges) {
  int e = blockIdx.x * blockDim.x + threadIdx.x;
  if (e < n_edges) norm[e] = dinv[row[e]] * ew[e] * dinv[col[e]];
}

// Pre-transpose weights so per-lane B-fragment pairs are contiguous:
//  W1T: 16x4, W1T[n][k] = W1[k][n], k=3 zero-padded (K 3->4)
//  W2T: 16x16, W2T[n][k] = W2[k][n]
__global__ void prep_wT_k(const float* __restrict__ W1, const float* __restrict__ W2,
                          float* __restrict__ W1T, float* __restrict__ W2T) {
  int t = threadIdx.x;  // one block of 256
  { int n0 = t >> 4, k = t & 15; W2T[n0 * 16 + k] = W2[k * 16 + n0]; }
  if (t < 64) { int n0 = t >> 2, k = t & 3; W1T[n0 * 4 + k] = (k < 3) ? W1[k * 16 + n0] : 0.0f; }
}

// ---- dense transforms via WMMA (one 16-node tile per wave) ----------------
// A 16x4 f32: lanes 0-15 hold K={0,1}, lanes 16-31 hold K={2,3} (2 VGPRs).
// D 16x16 f32: lanes 0-15 -> rows base+r (r=0..7) col=lane;
//              lanes 16-31 -> rows base+8+r, col=lane-16.

// h = x @ W1   (x: Nx3, W1T: 16x4 zero-padded)
__global__ void xw1_wmma_k(const float* __restrict__ x, const float* __restrict__ W1T,
                           float* __restrict__ h, int n) {
  int wave = (blockIdx.x * blockDim.x + threadIdx.x) >> 5;  // wave-uniform
  int lane = threadIdx.x & 31;
  int base = wave * 16;
  if (base >= n) return;                 // uniform exit: EXEC stays full
  int m  = lane & 15;
  int hi = lane >> 4;
  int node = base + m; if (node >= n) node = n - 1;  // clamp loads (tail only)

  v2f a, b;
  a.x = x[node * 3 + (hi << 1)];         // hi=0 -> K0, hi=1 -> K2
  float x1 = x[node * 3 + 1];
  a.y = hi ? 0.0f : x1;                  // K3 pad = 0
  b = ((const v2f*)W1T)[m * 2 + hi];     // contiguous pair, K3 pre-zeroed

  v8f c = {};
  c = __builtin_amdgcn_wmma_f32_16x16x4_f32(false, a, false, b, (short)0, c,
                                            false, false);
  int r0 = base + hi * 8;
  if (base + 16 <= n) {                  // uniform fast path: no per-lane guards
#pragma unroll
    for (int r = 0; r < 8; ++r) h[(r0 + r) * NF + m] = c[r];
  } else {
#pragma unroll
    for (int r = 0; r < 8; ++r) { int ri = r0 + r; if (ri < n) h[ri * NF + m] = c[r]; }
  }
}

// h2 = relu(acc) @ W2   (acc: Nx16, W2T: 16x16 transposed)
__global__ void gw2_wmma_k(const float* __restrict__ acc, const float* __restrict__ W2T,
                           float* __restrict__ h2, int n) {
  int wave = (blockIdx.x * blockDim.x + threadIdx.x) >> 5;
  int lane = threadIdx.x & 31;
  int base = wave * 16;
  if (base >= n) return;
  int m  = lane & 15;
  int hi = lane >> 4;
  int node = base + m; if (node >= n) node = n - 1;

  const v2f* accv = (const v2f*)acc;     // 8 v2f per node row
  const v2f* w2tv = (const v2f*)W2T;     // 8 v2f per output column
  v8f c = {};
#pragma unroll
  for (int k4 = 0; k4 < 4; ++k4) {       // chain 4x K=4 -> K=16
    v2f av = accv[node * 8 + k4 * 2 + hi];
    v2f a; a.x = fmaxf(av.x, 0.0f); a.y = fmaxf(av.y, 0.0f);   // fused relu
    v2f b = w2tv[m * 8 + k4 * 2 + hi];
    c = __builtin_amdgcn_wmma_f32_16x16x4_f32(false, a, false, b, (short)0, c,
                                              false, false);
  }
  int r0 = base + hi * 8;
  if (base + 16 <= n) {
#pragma unroll
    for (int r = 0; r < 8; ++r) h2[(r0 + r) * NF + m] = c[r];
  } else {
#pragma unroll
    for (int r = 0; r < 8; ++r) { int ri = r0 + r; if (ri < n) h2[ri * NF + m] = c[r]; }
  }
}

// ---- conv accumulate ------------------------------------------------------

// acc[i][j] = h[i][j]*dinv[i]^2 + bias[j]   (self-loop term + bias)
__global__ void init_acc_k(const float* __restrict__ h, const float* __restrict__ dinv,
                           const float* __restrict__ bias, float* __restrict__ acc,
                           int n) {
  int t = blockIdx.x * blockDim.x + threadIdx.x;
  if (t >= n * NF) return;
  int i = t >> 4, j = t & 15;
  float di = dinv[i];
  acc[t] = h[t] * di * di + bias[j];
}

// 16 lanes per edge: lane j adds h[row[e]][j]*norm[e] into acc[col[e]][j].
// Gather/atomic addresses are 16 consecutive floats -> one 64B segment/edge,
// all resolving in L2 (h is 6.4MB << 192MB).
__global__ void edge_agg_k(const int* __restrict__ row, const int* __restrict__ col,
                           const float* __restrict__ norm, const float* __restrict__ h,
                           float* __restrict__ acc, int n_edges) {
  int t = blockIdx.x * blockDim.x + threadIdx.x;
  int e = t >> 4;
  int j = t & 15;
  if (e >= n_edges) return;
  int r = row[e];
  int c = col[e];
  float v = h[r * NF + j] * norm[e];
  atomAddF(&acc[c * NF + j], v);
}

// ---- pooling + head -------------------------------------------------------

__global__ void zero_k(float* __restrict__ p, int n) {
  int i = blockIdx.x * blockDim.x + threadIdx.x;
  if (i < n) p[i] = 0.0f;
}

__global__ void pool_k(const float* __restrict__ acc2, const int* __restrict__ batch,
                       float* __restrict__ pooled, int n) {
  int t = blockIdx.x * blockDim.x + threadIdx.x;
  if (t >= n * NF) return;
  int i = t >> 4, j = t & 15;
  atomAddF(&pooled[batch[i] * NF + j], acc2[t]);
}

__global__ void head_k(const float* __restrict__ pooled, const float* __restrict__ Wl,
                       const float* __restrict__ bl, float* __restrict__ out,
                       int n_graphs) {
  int t = blockIdx.x * blockDim.x + threadIdx.x;
  if (t >= n_graphs * 7) return;
  int g = t / 7, o = t % 7;
  float s = bl[o];
#pragma unroll
  for (int j = 0; j < NF; ++j) s += pooled[g * NF + j] * Wl[j * 7 + o];
  out[t] = s;
}

// ---------------------------------------------------------------------------

extern "C" void kernel_launch(void* const* d_in, const int* in_sizes, int n_in,
                              void* d_out, int out_size, void* d_ws, size_t ws_size,
                              hipStream_t stream) {
  const int N = in_sizes[0] / 3;   // 100000 nodes
  const int E = in_sizes[2];       // 3200000 edges

  const float* x     = (const float*)d_in[0];
  const int*   ei    = (const int*)d_in[1];     // [2,E] int32 (JAX x64 off)
  const int*   row   = ei;
  const int*   col   = ei + E;
  const float* ew    = (const float*)d_in[2];
  const int*   batch = (const int*)d_in[3];
  const float* W1    = (const float*)d_in[4];
  const float* b1    = (const float*)d_in[5];
  const float* W2    = (const float*)d_in[6];
  const float* b2    = (const float*)d_in[7];
  const float* Wl    = (const float*)d_in[8];
  const float* bl    = (const float*)d_in[9];
  float*       out   = (float*)d_out;

  // workspace (floats):
  // dinv[N] | norm[E] | h[N*16] | acc[N*16] | acc2[N*16] | pooled[1024] | W1T[64] | W2T[256]
  float* F      = (float*)d_ws;
  float* dinv   = F;
  float* norm   = dinv + N;
  float* h      = norm + E;                   // holds x@W1, then reused for h2
  float* acc    = h + (size_t)N * NF;         // conv1 accumulator (pre-relu)
  float* acc2   = acc + (size_t)N * NF;       // conv2 accumulator
  float* pooled = acc2 + (size_t)N * NF;      // 64*16
  float* W1T    = pooled + 64 * NF;           // 16x4 (zero-padded)
  float* W2T    = W1T + 64;                   // 16x16

  const int B = 256;
  const int gN    = (N + B - 1) / B;
  const int gE    = (E + B - 1) / B;
  const int gNF   = (N * NF + B - 1) / B;
  const int gEF   = (E * NF + B - 1) / B;               // 200000 blocks
  const int gTile = (((N + 15) / 16) * 32 + B - 1) / B; // 16-node tile per wave

  // degree + normalization (shared by both convs) + weight transposes
  init_deg_k<<<gN, B, 0, stream>>>(dinv, N);
  prep_wT_k<<<1, B, 0, stream>>>(W1, W2, W1T, W2T);
  deg_scatter_k<<<gE, B, 0, stream>>>(col, ew, dinv, E);
  dinv_k<<<gN, B, 0, stream>>>(dinv, N);
  norm_k<<<gE, B, 0, stream>>>(row, col, ew, dinv, norm, E);

  // conv1: h = x@W1 ; acc = self + bias ; edge scatter-add
  xw1_wmma_k<<<gTile, B, 0, stream>>>(x, W1T, h, N);
  init_acc_k<<<gNF, B, 0, stream>>>(h, dinv, b1, acc, N);
  edge_agg_k<<<gEF, B, 0, stream>>>(row, col, norm, h, acc, E);

  // conv2: h2 = relu(acc)@W2 (into h) ; acc2 = self + bias ; edge scatter-add
  gw2_wmma_k<<<gTile, B, 0, stream>>>(acc, W2T, h, N);
  init_acc_k<<<gNF, B, 0, stream>>>(h, dinv, b2, acc2, N);
  edge_agg_k<<<gEF, B, 0, stream>>>(row, col, norm, h, acc2, E);

  // global add pool + linear head
  zero_k<<<(64 * NF + B - 1) / B, B, 0, stream>>>(pooled, 64 * NF);
  pool_k<<<gNF, B, 0, stream>>>(acc2, batch, pooled, N);
  head_k<<<(64 * 7 + B - 1) / B, B, 0, stream>>>(pooled, Wl, bl, out, 64);
}